// SignalPropagation_23192823398597
// MI455X (gfx1250) — compile-verified
//
#include <hip/hip_runtime.h>

// ---------------- problem constants (match reference) ----------------
constexpr int   kN   = 2048;    // nodes
constexpr int   kT   = 2048;    // time steps
constexpr int   kL   = 128;     // signal length; injected value = sig[t % (kL-1)]
constexpr float kMom = 0.999f;
constexpr float kEps = 1e-32f;

// ---------------- launch shape ----------------
// Persistent kernel: 128 blocks x 256 threads (8 wave32) = 1024 waves, trivially
// co-resident on an MI455X-class part -> grid spin-barrier is safe.
constexpr int kBlocks       = 128;
constexpr int kRowsPerBlock = kN / kBlocks;   // 16 = one WMMA M-tile per block
constexpr int kThreads      = 256;
constexpr int kWaves        = kThreads / 32;  // 8
constexpr int kKPerWave     = kN / kWaves;    // 256 K-depth per wave

typedef __attribute__((ext_vector_type(2))) float v2f;
typedef __attribute__((ext_vector_type(8))) float v8f;

// Zero S/Q ping-pong buffers and the grid-barrier words every call
// (d_ws is poisoned once and never re-poisoned; kernel must be deterministic).
__global__ void zero_ws_kernel(unsigned* __restrict__ p, int nwords) {
  int i = blockIdx.x * blockDim.x + threadIdx.x;
  if (i < nwords) p[i] = 0u;
}

// colsum[i] = sum_j A[j][i] + EPS ; rw[i] = rowsum of W = colsumA/(colsumA+EPS).
__global__ void colsum_kernel(const float* __restrict__ A,
                              float* __restrict__ cs,
                              float* __restrict__ rw) {
  int i = blockIdx.x * blockDim.x + threadIdx.x;
  if (i >= kN) return;
  float s = 0.0f;
#pragma unroll 8
  for (int j = 0; j < kN; ++j) s += A[(size_t)j * kN + i];  // coalesced across i
  cs[i] = s + kEps;
  rw[i] = s / (s + kEps);
}

// Persistent time-stepper. Per step:
//   Q = S_old + s_t + MOM*Q_old ; v = Q/cs  (redundant per block, into LDS)
//   y(block rows) = A[rows,:] @ v           (WMMA f32 16x16x4, broadcast-B)
//   S_new = S' + y - rw*Q ; out[row,t] = S' ; grid barrier.
__launch_bounds__(kThreads, 1)
__global__ void step_kernel(const float* __restrict__ A,
                            const float* __restrict__ sig,
                            const float* __restrict__ cs,
                            const float* __restrict__ rw,
                            float* __restrict__ Sbuf,       // 2*kN ping-pong
                            float* __restrict__ Qbuf,       // 2*kN ping-pong
                            unsigned* __restrict__ barrier, // [0]=count [1]=release
                            float* __restrict__ out)        // kN x kT row-major
{
  __shared__ __align__(16) float v_lds[kN];                 // 8 KB
  __shared__ __align__(16) float q_lds[kN];                 // 8 KB
  __shared__ __align__(16) float part[kWaves][kRowsPerBlock];

  const int tid   = threadIdx.x;
  const int lane  = tid & 31;
  const int wave  = tid >> 5;
  const int rbase = blockIdx.x * kRowsPerBlock;
  // WMMA f32 16x16x4 A-operand layout: lane holds M = lane%16, K = 2*(lane/16)+{0,1}
  const int mrow = rbase + (lane & 15);
  const int ksub = (lane >> 4) * 2;
  const float* __restrict__ arow = A + (size_t)mrow * kN;

  for (int t = 0; t < kT; ++t) {
    const int p = t & 1;
    const float* __restrict__ Sold = Sbuf + p * kN;
    const float* __restrict__ Qold = Qbuf + p * kN;
    float* __restrict__ Snew = Sbuf + (1 - p) * kN;
    float* __restrict__ Qnew = Qbuf + (1 - p) * kN;
    const float s_in = sig[t % (kL - 1)];

    // Phase 1: O(n) vector recurrence, recomputed redundantly per block.
    for (int j = tid; j < kN; j += kThreads) {
      float q = Sold[j] + s_in + kMom * Qold[j];
      q_lds[j] = q;
      v_lds[j] = q / cs[j];
    }
    __syncthreads();

    // Phase 2: y[rbase..rbase+15] partial over this wave's K range via WMMA.
    // B-operand broadcast: every N-column of B = v segment, so every column of
    // C holds the same matvec partial; column 0 is extracted below.
    v8f c0 = {0.f, 0.f, 0.f, 0.f, 0.f, 0.f, 0.f, 0.f};
    v8f c1 = {0.f, 0.f, 0.f, 0.f, 0.f, 0.f, 0.f, 0.f};
    int kb = wave * kKPerWave;
#pragma unroll 8
    for (int it = 0; it < kKPerWave / 8; ++it, kb += 8) {
      v2f a0 = *(const v2f*)(arow + kb + ksub);         // global b64, L2-resident A
      v2f b0 = *(const v2f*)(v_lds + kb + ksub);        // ds b64
      v2f a1 = *(const v2f*)(arow + kb + 4 + ksub);
      v2f b1 = *(const v2f*)(v_lds + kb + 4 + ksub);
      c0 = __builtin_amdgcn_wmma_f32_16x16x4_f32(false, a0, false, b0,
                                                 (short)0, c0, false, false);
      c1 = __builtin_amdgcn_wmma_f32_16x16x4_f32(false, a1, false, b1,
                                                 (short)0, c1, false, false);
    }
    c0 += c1;

    // Extract column N=0: lane 0 holds rows 0..7 (VGPR 0..7), lane 16 rows 8..15.
    if ((lane & 15) == 0) {
      const int h = (lane >> 4) * 8;
#pragma unroll
      for (int r = 0; r < 8; ++r) part[wave][h + r] = c0[r];
    }
    __syncthreads();

    // Phase 3: reduce wave partials, finalize this block's 16 rows.
    if (tid < kRowsPerBlock) {
      float y = 0.0f;
#pragma unroll
      for (int w = 0; w < kWaves; ++w) y += part[w][tid];
      const int r = rbase + tid;
      const float sp = Sold[r] + s_in;   // S'_t (the emitted response value)
      const float q  = q_lds[r];
      out[(size_t)r * kT + t] = sp;
      Snew[r] = sp + y - rw[r] * q;
      Qnew[r] = q;
    }

    // Phase 4: device-wide barrier (release value = t+1, monotone).
    __threadfence();
    __syncthreads();
    if (tid == 0) {
      const unsigned arrived = atomicAdd(&barrier[0], 1u);
      if (arrived == (unsigned)(kBlocks - 1)) {
        atomicExch(&barrier[0], 0u);   // reset before release; others still spin
        __threadfence();
        atomicExch(&barrier[1], (unsigned)(t + 1));
      } else {
        while (atomicAdd(&barrier[1], 0u) < (unsigned)(t + 1)) {
          __builtin_amdgcn_s_sleep(2);
        }
      }
    }
    __syncthreads();
    __threadfence();
  }
}

extern "C" void kernel_launch(void* const* d_in, const int* in_sizes, int n_in,
                              void* d_out, int out_size, void* d_ws, size_t ws_size,
                              hipStream_t stream) {
  const float* A   = (const float*)d_in[0];   // (kN, kN) fp32
  const float* sig = (const float*)d_in[1];   // (kL,)   fp32
  float* out = (float*)d_out;                 // (kN, kT) fp32

  // Workspace layout (fp32 words): cs[kN] | rw[kN] | S[2*kN] | Q[2*kN] | bar[2]
  float* ws = (float*)d_ws;
  float* cs = ws;
  float* rw = cs + kN;
  float* S  = rw + kN;
  float* Q  = S + 2 * kN;
  unsigned* bar = (unsigned*)(Q + 2 * kN);

  const int zero_words = 4 * kN + 2;  // S, Q ping-pong + barrier words
  zero_ws_kernel<<<(zero_words + 255) / 256, 256, 0, stream>>>((unsigned*)S,
                                                               zero_words);
  colsum_kernel<<<(kN + 255) / 256, 256, 0, stream>>>(A, cs, rw);
  step_kernel<<<kBlocks, kThreads, 0, stream>>>(A, sig, cs, rw, S, Q, bar, out);

  (void)in_sizes; (void)n_in; (void)out_size; (void)ws_size;
}